// ShardedEmbedIn_46076409151874
// MI455X (gfx1250) — compile-verified
//
#include <hip/hip_runtime.h>
#include <hip/hip_fp16.h>
#include <cstdint>

// ---------------------------------------------------------------------------
// Embedding gather via the CDNA5 Tensor Data Mover (gather mode).
//   out[row,:] = W[ids[row],:]   row in [0, B*S), hidden = 6144 fp16 = 12 KB
// Each 1-wave block:
//   TENSOR_LOAD_TO_LDS  (gather_mode, 4 x 32-bit row indices) -> 48 KB LDS tile
//   s_wait_tensorcnt 0
//   TENSOR_STORE_FROM_LDS (1-D 12288-dword tile) -> contiguous output
// ---------------------------------------------------------------------------

typedef uint32_t v4u __attribute__((ext_vector_type(4)));
typedef int32_t  v8i __attribute__((ext_vector_type(8)));
typedef int32_t  v4i __attribute__((ext_vector_type(4)));

#define HIDDEN        6144
#define ROW_BYTES     (HIDDEN * 2)          // 12288 B per embedding row
#define ROW_DW        (ROW_BYTES / 4)       // 3072 4-byte units (data_size=2)
#define ROWS_PER_BLK  4
#define TILE_BYTES    (ROWS_PER_BLK * ROW_BYTES)   // 49152 B LDS tile
#define TILE_DW       (ROWS_PER_BLK * ROW_DW)      // 12288 dwords
#define VOCAB         50432

// D# group 0: control | lds_addr | global_addr[56:0] | type=2
__device__ __forceinline__ v4u tdm_g0(uint32_t ctrl, uint32_t lds_byte_addr,
                                      uint64_t gaddr) {
  v4u g;
  g.x = ctrl;                                            // count=1 (+gather bits)
  g.y = lds_byte_addr;                                   // LDS byte address
  g.z = (uint32_t)gaddr;                                 // global_addr[31:0]
  g.w = ((uint32_t)(gaddr >> 32) & 0x01FFFFFFu)          // global_addr[56:32]
        | 0x80000000u;                                   // type = 2 ("image")
  return g;
}

// D# group 1 for the gather-mode LOAD: 2-D tensor [VOCAB rows x ROW_DW dwords],
// tile = ROWS_PER_BLK indexed rows x ROW_DW dwords.
__device__ __forceinline__ v8i tdm_g1_gather_load() {
  v8i g;
  g[0] = (int)(2u << 16);                         // wg_mask=0, data_size=2 (4B)
  g[1] = (int)((uint32_t)(ROW_DW & 0xFFFF) << 16);        // tensor_dim0 lo
  g[2] = (int)(((uint32_t)ROW_DW >> 16) |                 // tensor_dim0 hi
               (((uint32_t)VOCAB & 0xFFFF) << 16));       // tensor_dim1 lo
  g[3] = (int)(((uint32_t)VOCAB >> 16) |                  // tensor_dim1 hi
               ((uint32_t)ROW_DW << 16));                 // tile_dim0 = ROW_DW
  g[4] = (int)((uint32_t)ROWS_PER_BLK);           // tile_dim1 = #valid indices
  g[5] = (int)((uint32_t)ROW_DW);                 // tensor_dim0_stride lo
  g[6] = 0;                                       // stride hi / dim1_stride lo
  g[7] = 0;
  return g;
}

// D# group 1 for the STORE: plain 1-D tile of TILE_DW dwords.
__device__ __forceinline__ v8i tdm_g1_store() {
  v8i g;
  g[0] = (int)(2u << 16);                                 // data_size = 2 (4B)
  g[1] = (int)((uint32_t)(TILE_DW & 0xFFFF) << 16);       // tensor_dim0 lo
  g[2] = (int)(((uint32_t)TILE_DW >> 16) | (1u << 16));   // dim0 hi, tensor_dim1=1
  g[3] = (int)((uint32_t)(TILE_DW & 0xFFFF) << 16);       // tile_dim0 = TILE_DW
  g[4] = 0;                                               // tile_dim1=0 -> 1-D
  g[5] = (int)((uint32_t)TILE_DW);                        // tensor_dim0_stride
  g[6] = 0;
  g[7] = 0;
  return g;
}

__global__ __launch_bounds__(32)
void tdm_embed_gather(const int* __restrict__ ids,
                      const __half* __restrict__ W,
                      __half* __restrict__ out) {
  __shared__ __align__(128) unsigned char tile[TILE_BYTES];

  // Flat LDS aperture: addr[31:0] == LDS byte offset (ISA 10.2 aperture table).
  const uint32_t lds0 = (uint32_t)(uint64_t)(uintptr_t)tile;

  const int base = (int)blockIdx.x * ROWS_PER_BLK;

  // Row indices, forced wave-uniform (descriptor lives in SGPRs).
  const int v0 = __builtin_amdgcn_readfirstlane(ids[base + 0]);
  const int v1 = __builtin_amdgcn_readfirstlane(ids[base + 1]);
  const int v2 = __builtin_amdgcn_readfirstlane(ids[base + 2]);
  const int v3 = __builtin_amdgcn_readfirstlane(ids[base + 3]);

  const uint64_t wbase = (uint64_t)(uintptr_t)W;
  const uint64_t obase = (uint64_t)(uintptr_t)out + (uint64_t)base * ROW_BYTES;

  // ---- gather LOAD: 4 indexed rows of W -> LDS tile ----
  // ctrl: count=1, gather_index_size=1 (32-bit) bit30, gather_mode=1 bit31
  const v4u g0_ld = tdm_g0(0xC0000001u, lds0, wbase);
  const v8i g1_ld = tdm_g1_gather_load();
  const v4i gidx  = { v0, v1, v2, v3 };           // group 2: row_index_0..3
  const v4i zero4 = { 0, 0, 0, 0 };               // group 3: indices 4..7 unused
  const v8i zero8 = { 0, 0, 0, 0, 0, 0, 0, 0 };   // extra group (6-arg form)
  __builtin_amdgcn_tensor_load_to_lds(g0_ld, g1_ld, gidx, zero4, zero8, 0);

  // TDM ops from one wave are issue-ordered, but completion must be enforced
  // before the store reads the tile.
  __builtin_amdgcn_s_wait_tensorcnt(0);

  // ---- STORE: contiguous 48 KB LDS tile -> out[base..base+3] ----
  const v4u g0_st = tdm_g0(0x00000001u, lds0, obase);
  const v8i g1_st = tdm_g1_store();
  __builtin_amdgcn_tensor_store_from_lds(g0_st, g1_st, zero4, zero4, zero8, 0);

  __builtin_amdgcn_s_wait_tensorcnt(0);
}

extern "C" void kernel_launch(void* const* d_in, const int* in_sizes, int n_in,
                              void* d_out, int out_size, void* d_ws, size_t ws_size,
                              hipStream_t stream) {
  const int*    ids = (const int*)d_in[0];      // [B*S] int32
  const __half* W   = (const __half*)d_in[1];   // [VOCAB, HIDDEN] fp16
  __half*       out = (__half*)d_out;           // [B*S, HIDDEN] fp16

  const int rows   = in_sizes[0];               // B*S = 8192
  const int blocks = rows / ROWS_PER_BLK;       // 2048 one-wave workgroups

  tdm_embed_gather<<<dim3(blocks), dim3(32), 0, stream>>>(ids, W, out);
}